// MultiBoxLoss_3762391351809
// MI455X (gfx1250) — compile-verified
//
#include <hip/hip_runtime.h>
#include <hip/hip_bf16.h>
#include <math.h>

typedef unsigned char  u8;
typedef unsigned short u16;
typedef __attribute__((ext_vector_type(2))) float v2f;
typedef __attribute__((ext_vector_type(8))) float v8f;

#define NOBJ   16
#define PMAX   8732
#define SORTN  16384
#define TILE   64
#define ROWS   92          // padded LDS row stride (floats) for C=91
#define NEG_BIG (-3.402823466e38f)

// ---------------- CDNA5 async global->LDS helpers (gfx1250) ----------------
__device__ __forceinline__ void async_load_f32_to_lds(const float* gaddr, float* lds_ptr) {
    // Low 32 bits of a flat shared-aperture address are the LDS byte offset.
    unsigned int loff = (unsigned int)(unsigned long long)(uintptr_t)lds_ptr;
    unsigned long long ga = (unsigned long long)(uintptr_t)gaddr;
    asm volatile("global_load_async_to_lds_b32 %0, %1, off"
                 :: "v"(loff), "v"(ga) : "memory");
}
__device__ __forceinline__ void wait_async0() {
    asm volatile("s_wait_asynccnt 0x0" ::: "memory");
}

__device__ __forceinline__ float sl1(float x) {
    float a = fabsf(x);
    return (a < 1.0f) ? 0.5f * a * a : a - 0.5f;
}

// ---------------- K0: zero accumulators ----------------
__global__ __launch_bounds__(32) void k_init(float* accum) {
    if (threadIdx.x < 8) accum[threadIdx.x] = 0.0f;
}

// ---------------- K1: matching + loc loss (1 block per batch) ----------------
__global__ __launch_bounds__(256) void k_match(
    const float* __restrict__ plocs, const float* __restrict__ boxes,
    const int* __restrict__ labels, const float* __restrict__ priors,
    u8* __restrict__ flags, u8* __restrict__ labc,
    int* __restrict__ npos, float* __restrict__ accum, int P)
{
    __shared__ float s_bx[NOBJ * 4];     // xyxy
    __shared__ float s_area[NOBJ];
    __shared__ int   s_lab[NOBJ];
    __shared__ float s_ovl[PMAX];
    __shared__ u16   s_obj[PMAX];
    __shared__ u16   s_fp[NOBJ];
    __shared__ float red_f[256];
    __shared__ int   red_i[256];

    const int b = blockIdx.x, tid = threadIdx.x;
    if (tid < NOBJ) {
        size_t o = ((size_t)b * NOBJ + tid) * 4;
        float x = boxes[o + 0], y = boxes[o + 1], w = boxes[o + 2], h = boxes[o + 3];
        s_bx[tid*4+0] = x;      s_bx[tid*4+1] = y;
        s_bx[tid*4+2] = x + w;  s_bx[tid*4+3] = y + h;
        s_area[tid] = (s_bx[tid*4+2] - x) * (s_bx[tid*4+3] - y);
        s_lab[tid]  = labels[b * NOBJ + tid];
    }
    __syncthreads();

    // Phase A: per-prior best object; per-object best prior (thread-local)
    float bo[NOBJ]; int bp[NOBJ];
#pragma unroll
    for (int i = 0; i < NOBJ; ++i) { bo[i] = -1.0f; bp[i] = 0x7fffffff; }

    for (int p = tid; p < P; p += 256) {
        float pcx = priors[p*4+0], pcy = priors[p*4+1];
        float pw  = priors[p*4+2], ph  = priors[p*4+3];
        float px1 = pcx - pw*0.5f, py1 = pcy - ph*0.5f;
        float px2 = pcx + pw*0.5f, py2 = pcy + ph*0.5f;
        float pa  = (px2 - px1) * (py2 - py1);
        float best = -1.0f; int bi = 0;
#pragma unroll
        for (int i = 0; i < NOBJ; ++i) {
            float ax1 = s_bx[i*4+0], ay1 = s_bx[i*4+1];
            float ax2 = s_bx[i*4+2], ay2 = s_bx[i*4+3];
            float iw = fmaxf(fminf(ax2, px2) - fmaxf(ax1, px1), 0.0f);
            float ih = fmaxf(fminf(ay2, py2) - fmaxf(ay1, py1), 0.0f);
            float inter = iw * ih;
            float iou = inter / (s_area[i] + pa - inter);
            if (iou > best) { best = iou; bi = i; }
            if (iou > bo[i]) { bo[i] = iou; bp[i] = p; }
        }
        s_ovl[p] = best; s_obj[p] = (u16)bi;
    }
    __syncthreads();

    // Phase B: reduce per-object best prior across block (tie -> smaller p)
    for (int i = 0; i < NOBJ; ++i) {
        red_f[tid] = bo[i]; red_i[tid] = bp[i];
        __syncthreads();
        for (int st = 128; st > 0; st >>= 1) {
            if (tid < st) {
                float f2 = red_f[tid+st]; int p2 = red_i[tid+st];
                if (f2 > red_f[tid] || (f2 == red_f[tid] && p2 < red_i[tid])) {
                    red_f[tid] = f2; red_i[tid] = p2;
                }
            }
            __syncthreads();
        }
        if (tid == 0) s_fp[i] = (u16)red_i[0];
        __syncthreads();
    }

    // Phase C: forced assignment (sequential i -> last-writer-wins on dup priors)
    if (tid == 0) {
        for (int i = 0; i < NOBJ; ++i) {
            int p = s_fp[i];
            s_obj[p] = (u16)i;
            s_ovl[p] = 1.0f;
        }
    }
    __syncthreads();

    // Phase D: labels/flags + loc loss for positives
    float locsum = 0.0f; int poscnt = 0;
    size_t bP = (size_t)b * P;
    for (int p = tid; p < P; p += 256) {
        float ovl = s_ovl[p];
        int oi = s_obj[p];
        int lab = s_lab[oi];
        if (ovl < 0.5f) lab = 0;
        bool ign = (ovl > 0.4f) && (ovl < 0.5f);
        bool pos = (lab > 0) && !ign;
        flags[bP + p] = pos ? (u8)1 : (ign ? (u8)2 : (u8)0);
        labc[bP + p]  = (u8)lab;
        if (pos) {
            float ax1 = s_bx[oi*4+0], ay1 = s_bx[oi*4+1];
            float ax2 = s_bx[oi*4+2], ay2 = s_bx[oi*4+3];
            float ocx = (ax1 + ax2) * 0.5f, ocy = (ay1 + ay2) * 0.5f;
            float ow  = ax2 - ax1,          oh  = ay2 - ay1;
            float pcx = priors[p*4+0], pcy = priors[p*4+1];
            float pw  = priors[p*4+2], ph  = priors[p*4+3];
            float g0 = (ocx - pcx) / (pw * 0.1f);
            float g1 = (ocy - pcy) / (ph * 0.1f);
            float g2 = logf(fabsf(ow / pw)) * 5.0f;
            float g3 = logf(fabsf(oh / ph)) * 5.0f;
            const float* lp = &plocs[(bP + p) * 4];
            locsum += sl1(lp[0]-g0) + sl1(lp[1]-g1) + sl1(lp[2]-g2) + sl1(lp[3]-g3);
            poscnt++;
        }
    }
    red_f[tid] = locsum; red_i[tid] = poscnt;
    __syncthreads();
    for (int st = 128; st > 0; st >>= 1) {
        if (tid < st) { red_f[tid] += red_f[tid+st]; red_i[tid] += red_i[tid+st]; }
        __syncthreads();
    }
    if (tid == 0) { atomicAdd(&accum[0], red_f[0]); npos[b] = red_i[0]; }
}

// ---------------- K2: per-prior cross entropy (bandwidth kernel) ----------------
// Tile of TILE priors x C classes staged to LDS via async global->LDS copies.
// Pad columns [C,ROWS) and rows [n,TILE) are pre-filled with NEG_BIG so the
// WMMA loop needs no bounds guards (no exec-divergent ds_loads).
// Pass 1: wave32-shuffle row max. Pass 2: sum-of-exp via V_WMMA_F32_16X16X4_F32
// against an all-ones B matrix (16 priors per wave, 23 accumulating WMMAs).
__global__ __launch_bounds__(256) void k_ce(
    const float* __restrict__ scores, const u8* __restrict__ flags,
    const u8* __restrict__ labc, float* __restrict__ ceneg,
    float* __restrict__ accum, int P, int C)
{
    __shared__ float s_sc[TILE * ROWS];
    __shared__ float s_max[TILE];
    __shared__ float s_red[256];

    const int b  = blockIdx.y;
    const int p0 = blockIdx.x * TILE;
    const int n  = min(TILE, P - p0);
    const int tid = threadIdx.x;

    // Async-stage the valid region: rows [0,n) x cols [0,C)
    const float* base = scores + ((size_t)b * P + p0) * C;
    const int tot = n * C;
    for (int d = tid; d < tot; d += 256) {
        int row = d / C;
        int col = d - row * C;
        async_load_f32_to_lds(base + d, &s_sc[row * ROWS + col]);
    }
    // Fill pad columns [C,ROWS) of valid rows (disjoint from async target)
    {
        const int padw = ROWS - C;                       // 1 for C=91
        for (int i = tid; i < n * padw; i += 256) {
            int r = i / padw, c = C + (i - r * padw);
            s_sc[r * ROWS + c] = NEG_BIG;
        }
        // Fill whole rows [n,TILE) (partial last tile only)
        for (int i = tid; i < (TILE - n) * ROWS; i += 256) {
            int r = n + i / ROWS, c = i % ROWS;
            s_sc[r * ROWS + c] = NEG_BIG;
        }
        if (tid >= n && tid < TILE) s_max[tid] = 0.0f;
    }
    wait_async0();          // each wave waits for its own async copies
    __syncthreads();        // block-wide: LDS tile (async + fills) visible

    const int wave = tid >> 5, lane = tid & 31;

    // Pass 1: per-prior max (one wave per prior, shuffle reduction)
    for (int r = wave; r < n; r += 8) {
        const float* rowp = &s_sc[r * ROWS];
        float x0 = rowp[lane];
        float x1 = (lane + 32 < C) ? rowp[lane + 32] : NEG_BIG;
        float x2 = (lane + 64 < C) ? rowp[lane + 64] : NEG_BIG;
        float m = fmaxf(x0, fmaxf(x1, x2));
        for (int o = 16; o > 0; o >>= 1) m = fmaxf(m, __shfl_xor(m, o, 32));
        if (lane == 0) s_max[r] = m;
    }
    __syncthreads();

    float posacc = 0.0f;
#if __has_builtin(__builtin_amdgcn_wmma_f32_16x16x4_f32)
    // Pass 2: sum of exp via matrix pipe. A = exp(tile rows - max), B = ones.
    // A(16x4) layout: lane l and l+16 both hold row (l&15); K pairs {0,1}/{2,3}.
    // All loads are in-bounds by construction (pads = NEG_BIG -> exp = 0).
    const int groups = (n + 15) >> 4;
    for (int g = wave; g < groups; g += 8) {
        const int r0 = g << 4;
        const int row = r0 + (lane & 15);
        const float m = s_max[row];
        const float* rowp = &s_sc[row * ROWS + ((lane >= 16) ? 2 : 0)];
        v2f ones; ones.x = 1.0f; ones.y = 1.0f;
        v8f c = {};
        for (int k = 0; k < C; k += 4) {
            float x0 = rowp[k];
            float x1 = rowp[k + 1];
            v2f a; a.x = expf(x0 - m); a.y = expf(x1 - m);
            c = __builtin_amdgcn_wmma_f32_16x16x4_f32(
                    false, a, false, ones, (short)0, c, false, false);
        }
        // D layout: lanes 0-15 vgpr j = row r0+j; lanes 16-31 vgpr j = row r0+j+8.
        float e = 0.0f; int row2 = -1;
        if (lane < 8)                       { e = c[lane];      row2 = r0 + lane; }
        else if (lane >= 16 && lane < 24)   { e = c[lane - 16]; row2 = r0 + (lane - 16) + 8; }
        if (row2 >= 0 && row2 < n) {
            size_t idx = (size_t)b * P + p0 + row2;
            int lab = labc[idx];
            float ce = logf(e) + s_max[row2] - s_sc[row2 * ROWS + lab];
            int fl = flags[idx];
            ceneg[idx] = (fl == 0) ? ce : 0.0f;
            if (fl == 1) posacc += ce;
        }
    }
#else
    // Fallback: shuffle-based sum of exp (one wave per prior)
    for (int r = wave; r < n; r += 8) {
        const float* rowp = &s_sc[r * ROWS];
        float m = s_max[r];
        float x0 = rowp[lane];
        float x1 = (lane + 32 < C) ? rowp[lane + 32] : NEG_BIG;
        float x2 = (lane + 64 < C) ? rowp[lane + 64] : NEG_BIG;
        float e = expf(x0 - m) + expf(x1 - m) + expf(x2 - m);
        for (int o = 16; o > 0; o >>= 1) e += __shfl_xor(e, o, 32);
        size_t idx = (size_t)b * P + p0 + r;
        int lab = labc[idx];
        float ce = logf(e) + m - rowp[lab];
        if (lane == 0) {
            int fl = flags[idx];
            ceneg[idx] = (fl == 0) ? ce : 0.0f;
            if (fl == 1) posacc += ce;
        }
    }
#endif
    s_red[tid] = posacc;
    __syncthreads();
    for (int st = 128; st > 0; st >>= 1) {
        if (tid < st) s_red[tid] += s_red[tid+st];
        __syncthreads();
    }
    if (tid == 0 && s_red[0] != 0.0f) atomicAdd(&accum[1], s_red[0]);
}

// ---------------- K3: hard-negative top-K sum via bitonic sort ----------------
__global__ __launch_bounds__(1024) void k_topk(
    const float* __restrict__ ceneg, const int* __restrict__ npos,
    float* __restrict__ accum, int P)
{
    __shared__ float s[SORTN];
    const int b = blockIdx.x, tid = threadIdx.x;
    for (int i = tid; i < SORTN; i += 1024)
        s[i] = (i < P) ? ceneg[(size_t)b * P + i] : -1.0f;   // ce >= 0, pad sorts last
    __syncthreads();

    for (int k = 2; k <= SORTN; k <<= 1) {
        for (int j = k >> 1; j > 0; j >>= 1) {
            for (int i = tid; i < SORTN; i += 1024) {
                int ix = i ^ j;
                if (ix > i) {
                    float a = s[i], c = s[ix];
                    bool desc = ((i & k) == 0);
                    if (desc ? (a < c) : (a > c)) { s[i] = c; s[ix] = a; }
                }
            }
            __syncthreads();
        }
    }

    int K = npos[b] * 3;          // NEG_POS_RATIO
    if (K > P) K = P;
    float acc = 0.0f;
    for (int i = tid; i < K; i += 1024) acc += s[i];
    __syncthreads();
    s[tid] = acc;
    __syncthreads();
    for (int st = 512; st > 0; st >>= 1) {
        if (tid < st) s[tid] += s[tid+st];
        __syncthreads();
    }
    if (tid == 0) atomicAdd(&accum[2], s[0]);
}

// ---------------- K4: finalize ----------------
__global__ __launch_bounds__(64) void k_final(
    const int* __restrict__ npos, const float* __restrict__ accum,
    float* __restrict__ out, int B)
{
    __shared__ int s[64];
    const int tid = threadIdx.x;
    s[tid] = (tid < B) ? npos[tid] : 0;
    __syncthreads();
    for (int st = 32; st > 0; st >>= 1) {
        if (tid < st) s[tid] += s[tid+st];
        __syncthreads();
    }
    if (tid == 0) {
        float npt  = (float)s[0];
        float conf = (accum[1] + accum[2]) / npt;
        float loc  = accum[0] / (npt * 4.0f);
        out[0] = conf + 1.0f * loc;   // total (ALPHA = 1)
        out[1] = conf;
        out[2] = loc;
    }
}

// ---------------- host side ----------------
extern "C" void kernel_launch(void* const* d_in, const int* in_sizes, int n_in,
                              void* d_out, int out_size, void* d_ws, size_t ws_size,
                              hipStream_t stream) {
    const float* plocs  = (const float*)d_in[0];
    const float* scores = (const float*)d_in[1];
    const float* boxes  = (const float*)d_in[2];
    const int*   labels = (const int*)d_in[3];
    const float* priors = (const float*)d_in[4];

    const int P = in_sizes[4] / 4;                       // 8732
    const int B = in_sizes[3] / NOBJ;                    // 64
    const int C = (int)((long long)in_sizes[1] / ((long long)B * P)); // 91

    char* ws = (char*)d_ws;
    float* accum = (float*)ws;                           // [0]=loc_sum [1]=pos_sum [2]=hard_sum
    int*   npos  = (int*)(ws + 256);                     // [B]
    size_t BP  = (size_t)B * P;
    size_t BPa = (BP + 255) & ~(size_t)255;
    u8* flags = (u8*)(ws + 512);
    u8* labc  = flags + BPa;
    float* ceneg = (float*)(ws + 512 + 2 * BPa);

    k_init <<<1, 32, 0, stream>>>(accum);
    k_match<<<B, 256, 0, stream>>>(plocs, boxes, labels, priors, flags, labc, npos, accum, P);
    dim3 g2((P + TILE - 1) / TILE, B);
    k_ce   <<<g2, 256, 0, stream>>>(scores, flags, labc, ceneg, accum, P, C);
    k_topk <<<B, 1024, 0, stream>>>(ceneg, npos, accum, P);
    k_final<<<1, 64, 0, stream>>>(npos, accum, (float*)d_out, B);
}